// OpticalFlowEstimator_17128329576737
// MI455X (gfx1250) — compile-verified
//
#include <hip/hip_runtime.h>

// Optical-flow block matching, fused single kernel for MI455X (gfx1250).
//
// Roofline: min DRAM traffic = 2*50.3MB in + 33.5MB out = 134MB -> 5.8us @ 23.3TB/s.
// Compute: 4.19M px * 81 shifts * ~5 VALU = ~1.7G ops -> VALU-bound (~15-25us),
// so: single fused kernel (DRAM touched once), LDS-staged target tile via the
// CDNA5 async global->LDS copy engine, register sliding window (4 px/thread)
// to cut LDS reads 12x, and a packed shift-code so each shift costs 5 VALU ops.
// No WMMA: per-shift correlation is a single scalar abs-diff (no K dimension),
// so matrix ops cannot express this; the gfx1250-specific win is the
// global_load_async_to_lds_* / s_wait_asynccnt path.

namespace {
constexpr int Bn = 16, Cn = 3, Hn = 512, Wn = 512;
constexpr int SR  = 4;                   // search range
constexpr int TW  = 128;                 // tile width in pixels
constexpr int THt = 8;                   // tile height in pixels
constexpr int HALO_W = TW + 2 * SR;      // 136
constexpr int HALO_H = THt + 2 * SR;     // 16
constexpr int NTHREADS = 256;            // 8 wave32
}

__global__ __launch_bounds__(NTHREADS)
void flow_kernel(const float* __restrict__ src,
                 const float* __restrict__ tgt,
                 float* __restrict__ out)
{
    // 3 channels of the target halo tile; ch[0] is grayscaled in place later.
    __shared__ __align__(16) float ch[3][HALO_H][HALO_W];   // 26112 B

    const int tid = threadIdx.x;
    const int x0t = blockIdx.x * TW;
    const int y0t = blockIdx.y * THt;
    const int b   = blockIdx.z;
    const size_t imgHW = (size_t)Hn * Wn;

    // ---- Phase 1: async-copy target halo tile (3 channels) global -> LDS ----
    // Edge padding == clamped indexing. Per-wave ASYNCcnt stays <= 27 (< 63).
    const float* tbase = tgt + (size_t)b * Cn * imgHW;
    for (int idx = tid; idx < HALO_H * HALO_W; idx += NTHREADS) {
        const int r = idx / HALO_W;
        const int c = idx - r * HALO_W;
        int gy = y0t + r - SR; gy = gy < 0 ? 0 : (gy > Hn - 1 ? Hn - 1 : gy);
        int gx = x0t + c - SR; gx = gx < 0 ? 0 : (gx > Wn - 1 ? Wn - 1 : gx);
        const size_t off = (size_t)gy * Wn + gx;
#pragma unroll
        for (int k = 0; k < 3; ++k) {
            unsigned long long ga =
                (unsigned long long)(size_t)(tbase + (size_t)k * imgHW + off);
            // low 32 bits of a flat LDS address are the LDS byte offset
            unsigned ldsa = (unsigned)(size_t)(&ch[k][0][0] + idx);
            asm volatile("global_load_async_to_lds_b32 %0, %1, off"
                         :: "v"(ldsa), "v"(ga) : "memory");
        }
    }

    // ---- Phase 1b (overlaps async): source pixels -> regs, grayscale ----
    const int tx = tid & 31;     // 32 threads cover 128 px, 4 each
    const int ty = tid >> 5;     // 8 rows
    const int x0 = x0t + tx * 4;
    const int y  = y0t + ty;
    const float* sbase = src + (size_t)b * Cn * imgHW + (size_t)y * Wn + x0;
    const float4 s0 = *reinterpret_cast<const float4*>(sbase);
    const float4 s1 = *reinterpret_cast<const float4*>(sbase + imgHW);
    const float4 s2 = *reinterpret_cast<const float4*>(sbase + 2 * imgHW);
    float sg[4];
    sg[0] = (s0.x + s1.x + s2.x) * (1.0f / 3.0f);
    sg[1] = (s0.y + s1.y + s2.y) * (1.0f / 3.0f);
    sg[2] = (s0.z + s1.z + s2.z) * (1.0f / 3.0f);
    sg[3] = (s0.w + s1.w + s2.w) * (1.0f / 3.0f);

    asm volatile("s_wait_asynccnt 0x0" ::: "memory");
    __syncthreads();

    // ---- Phase 2: grayscale target tile in place (into ch[0]) ----
    {
        float*       c0 = &ch[0][0][0];
        const float* c1 = &ch[1][0][0];
        const float* c2 = &ch[2][0][0];
        for (int idx = tid; idx < HALO_H * HALO_W; idx += NTHREADS)
            c0[idx] = (c0[idx] + c1[idx] + c2[idx]) * (1.0f / 3.0f);
    }
    __syncthreads();

    // ---- Phase 3: 81-shift search, register sliding window ----
    // Reference order: dy outer (-4..4), dx inner (-4..4), strict improvement.
    float bestad[4] = {3.4e38f, 3.4e38f, 3.4e38f, 3.4e38f};
    float code[4]   = {0.f, 0.f, 0.f, 0.f};
    const float* gr = &ch[0][0][0];
#pragma unroll
    for (int dyi = 0; dyi < 9; ++dyi) {
        // LDS row (ty+dyi), cols [tx*4 .. tx*4+11]: 3 aligned ds_load_b128
        const float* row = gr + (ty + dyi) * HALO_W + tx * 4;
        const float4 a  = *reinterpret_cast<const float4*>(row);
        const float4 bq = *reinterpret_cast<const float4*>(row + 4);
        const float4 cq = *reinterpret_cast<const float4*>(row + 8);
        const float w[12] = {a.x, a.y, a.z, a.w,
                             bq.x, bq.y, bq.z, bq.w,
                             cq.x, cq.y, cq.z, cq.w};
#pragma unroll
        for (int dxi = 0; dxi < 9; ++dxi) {
            const float fc = (float)(dyi * 9 + dxi);
#pragma unroll
            for (int p = 0; p < 4; ++p) {
                const float d  = sg[p] - w[dxi + p];
                const float ad = __builtin_fabsf(d);       // corr = -ad
                const bool better = ad < bestad[p];        // == corr > best
                bestad[p] = better ? ad : bestad[p];
                code[p]   = better ? fc : code[p];
            }
        }
    }

    // ---- Epilogue: decode shift code, store flow_x / flow_y ----
    float fx[4], fy[4];
#pragma unroll
    for (int p = 0; p < 4; ++p) {
        const int ic = (int)code[p];
        const int q  = ic / 9;           // dy index 0..8
        const int rm = ic - q * 9;       // dx index 0..8
        fx[p] = (float)(rm - SR);
        fy[p] = (float)(q - SR);
    }
    float* outx = out + (size_t)b * 2 * imgHW + (size_t)y * Wn + x0;
    float* outy = outx + imgHW;
    *reinterpret_cast<float4*>(outx) = make_float4(fx[0], fx[1], fx[2], fx[3]);
    *reinterpret_cast<float4*>(outy) = make_float4(fy[0], fy[1], fy[2], fy[3]);
}

extern "C" void kernel_launch(void* const* d_in, const int* in_sizes, int n_in,
                              void* d_out, int out_size, void* d_ws, size_t ws_size,
                              hipStream_t stream) {
    (void)in_sizes; (void)n_in; (void)d_ws; (void)ws_size; (void)out_size;
    const float* src = (const float*)d_in[0];
    const float* tgt = (const float*)d_in[1];
    float* out = (float*)d_out;   // (16, 2, 512, 512) float32
    dim3 grid(Wn / TW, Hn / THt, Bn);   // 4 x 64 x 16 = 4096 blocks
    hipLaunchKernelGGL(flow_kernel, grid, dim3(NTHREADS), 0, stream,
                       src, tgt, out);
}